// PatchDirectionalConv2D_23218593202414
// MI455X (gfx1250) — compile-verified
//
#include <hip/hip_runtime.h>
#include <cstdint>

typedef uint32_t u32;
typedef u32 u32x4 __attribute__((ext_vector_type(4)));
typedef u32 u32x8 __attribute__((ext_vector_type(8)));

#define IMG_W 768
#define IMG_H 768
#define NB    16
#define TILE_PW 32            // patches per tile (x)
#define TILE_PH 8             // patches per tile (y)
#define TPW (TILE_PW * 3)     // 96 pixels
#define TPH (TILE_PH * 3)     // 24 pixels
#define LDS_ELEMS (TPW * TPH) // 2304 floats
#define LDS_BYTES (LDS_ELEMS * 4)

// ---- Tensor DMA Descriptor (D#) builders, per CDNA5 ISA ----
// Group 0 (4 SGPRs): [1:0]=count=1, [63:32]=lds_addr, [120:64]=global_addr, [127:126]=type=2
__device__ __forceinline__ u32x4 tdm_group0(u32 lds_addr, const void* gaddr) {
  uint64_t ga = (uint64_t)(uintptr_t)gaddr;
  u32x4 g;
  g[0] = 1u;                                                   // count=1, user descriptor
  g[1] = lds_addr;                                             // LDS byte address
  g[2] = (u32)ga;                                              // global_addr[31:0]
  g[3] = (u32)((ga >> 32) & 0x01FFFFFFull) | (2u << 30);       // global_addr[56:32], type=2
  return g;
}

// Group 1 (8 SGPRs): data_size=4B, 2D tile 96x24, row stride 768, workgroup_mask=0
__device__ __forceinline__ u32x8 tdm_group1() {
  const u32 td0 = IMG_W;            // tensor_dim0
  const u32 td1 = IMG_H * NB;       // tensor_dim1 (B folded into rows)
  u32x8 g;
  g[0] = (2u << 16);                                  // mask=0, data_size=2 (4 bytes)
  g[1] = (td0 & 0xFFFFu) << 16;                       // abar_addr=0 | tensor_dim0[15:0]
  g[2] = (td0 >> 16) | ((td1 & 0xFFFFu) << 16);       // tensor_dim0[31:16] | tensor_dim1[15:0]
  g[3] = (td1 >> 16) | ((u32)TPW << 16);              // tensor_dim1[31:16] | tile_dim0=96
  g[4] = (u32)TPH;                                    // tile_dim1=24 | tile_dim2=0 (2D)
  g[5] = (u32)IMG_W;                                  // tensor_dim0_stride[31:0]
  g[6] = 0u;                                          // stride0 hi | stride1 lo (unused, 2D)
  g[7] = 0u;
  return g;
}

__global__ __launch_bounds__(256) void patchdirconv_kernel(
    const float* __restrict__ Yg, const float* __restrict__ Ug,
    const float* __restrict__ Vg, float* __restrict__ Og) {
  extern __shared__ float smem[];
  float* sY = smem;
  float* sU = smem + LDS_ELEMS;
  float* sV = smem + 2 * LDS_ELEMS;

  const int tx = blockIdx.x;   // 0..7
  const int ty = blockIdx.y;   // 0..31
  const int bb = blockIdx.z;   // 0..15
  const size_t tile_off =
      ((size_t)(bb * IMG_H + ty * TPH)) * IMG_W + (size_t)(tx * TPW);

  // ---- Wave 0 issues three TDM tile loads (EXEC-independent, TENSORcnt-tracked) ----
  if (threadIdx.x < 32u) {
    const u32 lds0 = (u32)(uintptr_t)smem;   // flat addr low 32 bits == LDS byte offset
    const u32x8 g1 = tdm_group1();
    const u32x4 gy = tdm_group0(lds0,                 Yg + tile_off);
    const u32x4 gu = tdm_group0(lds0 + LDS_BYTES,     Ug + tile_off);
    const u32x4 gv = tdm_group0(lds0 + 2 * LDS_BYTES, Vg + tile_off);
    asm volatile("tensor_load_to_lds %0, %1" :: "s"(gy), "s"(g1) : "memory");
    asm volatile("tensor_load_to_lds %0, %1" :: "s"(gu), "s"(g1) : "memory");
    asm volatile("tensor_load_to_lds %0, %1" :: "s"(gv), "s"(g1) : "memory");
    __builtin_amdgcn_s_wait_tensorcnt(0);
  }
  __syncthreads();

  // ---- One thread per 3x3 patch ----
  const int px = (int)(threadIdx.x & 31u);
  const int py = (int)(threadIdx.x >> 5);
  const int lr = py * 3, lc = px * 3;

  float Yv[3][3];
  float us = 0.f, vs = 0.f;
#pragma unroll
  for (int r = 0; r < 3; ++r)
#pragma unroll
    for (int c = 0; c < 3; ++c) {
      const int idx = (lr + r) * TPW + (lc + c);
      Yv[r][c] = sY[idx];
      us += sU[idx];
      vs += sV[idx];
    }

  // cos/sin of atan2(v_mean,u_mean) == normalize(us,vs); atan2(0,0)=0 -> (1,0)
  const float rr  = us * us + vs * vs;
  const float inv = rsqrtf(rr);
  const bool  zz  = !(rr > 0.f);
  const float ct  = zz ? 1.f : us * inv;
  const float st  = zz ? 0.f : vs * inv;

  // GRID_X[a][b]=(b-2)/3, GRID_Y[a][b]=(a-2)/3, DECAY=exp(-(dx^2+dy^2)/1.5)
  const float G[5] = {-2.f / 3.f, -1.f / 3.f, 0.f, 1.f / 3.f, 2.f / 3.f};
  const float DEC[5][5] = {
      {0.55289201f, 0.69047862f, 0.74356718f, 0.69047862f, 0.55289201f},
      {0.69047862f, 0.86230340f, 0.92860292f, 0.86230340f, 0.69047862f},
      {0.74356718f, 0.92860292f, 1.00000000f, 0.92860292f, 0.74356718f},
      {0.69047862f, 0.86230340f, 0.92860292f, 0.86230340f, 0.69047862f},
      {0.55289201f, 0.69047862f, 0.74356718f, 0.69047862f, 0.55289201f}};

  float w[5][5];
  float wsum = 0.f;
#pragma unroll
  for (int a = 0; a < 5; ++a)
#pragma unroll
    for (int b2 = 0; b2 < 5; ++b2) {
      float t = fmaf(ct, G[b2], st * G[a]) * DEC[a][b2];
      t = fmaxf(t, 0.f);
      w[a][b2] = t;
      wsum += t;
    }
  const float sc = 1.f / (wsum + 1e-6f);   // defer normalization to the store

  // Factored conv: out[i][j] = sum_{r,c} (row-bucket of col-bucket of w) * Y[r][c]
  // Column buckets per output col j: b-ranges with clamp(j+b-2,0,2)==c
  const int CLO[3][3] = {{0, 3, 4}, {0, 2, 3}, {0, 1, 2}};
  const int CHI[3][3] = {{2, 3, 4}, {1, 2, 4}, {0, 1, 4}};

  float out[3][3] = {{0.f, 0.f, 0.f}, {0.f, 0.f, 0.f}, {0.f, 0.f, 0.f}};
#pragma unroll
  for (int j = 0; j < 3; ++j)
#pragma unroll
    for (int c = 0; c < 3; ++c) {
      float cs[5];
#pragma unroll
      for (int a = 0; a < 5; ++a) {
        float s = w[a][CLO[j][c]];
#pragma unroll
        for (int b2 = CLO[j][c] + 1; b2 <= CHI[j][c]; ++b2) s += w[a][b2];
        cs[a] = s;
      }
      // Row buckets: i=0 -> {012,3,4}; i=1 -> {01,2,34}; i=2 -> {0,1,234}
      const float s01 = cs[0] + cs[1];
      const float s012 = s01 + cs[2];
      const float s34 = cs[3] + cs[4];
      const float s234 = cs[2] + s34;
      const float y0 = Yv[0][c], y1 = Yv[1][c], y2 = Yv[2][c];
      out[0][j] += s012 * y0 + cs[3] * y1 + cs[4] * y2;
      out[1][j] += s01 * y0 + cs[2] * y1 + s34 * y2;
      out[2][j] += cs[0] * y0 + cs[1] * y1 + s234 * y2;
    }

  float* op = Og + tile_off + (size_t)lr * IMG_W + lc;
#pragma unroll
  for (int i = 0; i < 3; ++i)
#pragma unroll
    for (int j = 0; j < 3; ++j) op[i * IMG_W + j] = out[i][j] * sc;
}

extern "C" void kernel_launch(void* const* d_in, const int* in_sizes, int n_in,
                              void* d_out, int out_size, void* d_ws, size_t ws_size,
                              hipStream_t stream) {
  (void)in_sizes; (void)n_in; (void)out_size; (void)d_ws; (void)ws_size;
  const float* Y = (const float*)d_in[0];
  const float* U = (const float*)d_in[1];
  const float* V = (const float*)d_in[2];
  float* O = (float*)d_out;
  dim3 grid(IMG_W / TPW, IMG_H / TPH, NB);   // (8, 32, 16)
  patchdirconv_kernel<<<grid, dim3(256), 3 * LDS_BYTES, stream>>>(Y, U, V, O);
}